// Folded_Encoder_11252814315788
// MI455X (gfx1250) — compile-verified
//
#include <hip/hip_runtime.h>

#define D 64
#define TWO_D 128

typedef __attribute__((ext_vector_type(2))) float v2f;
typedef __attribute__((ext_vector_type(8))) float v8f;

__global__ __launch_bounds__(512)
void folded_encoder_fused(const int* __restrict__ nodes,
                          const int* __restrict__ node_seq,
                          const float* __restrict__ uv2e,
                          const float* __restrict__ feat,
                          const float* __restrict__ W1,
                          const float* __restrict__ b1,
                          float* __restrict__ out,
                          int B, int S) {
    // 16 rows x 128 cols of "combined" staged for the WMMA phase (8 KB)
    __shared__ float comb[16 * TWO_D];

    const int wave = threadIdx.x >> 5;   // 0..15
    const int lane = threadIdx.x & 31;
    const int row  = blockIdx.x * 16 + wave;

    // ---------- Phase 1: per-row attention (one wave32 per row) ----------
    if (row < B) {
        const int nid = nodes[row];
        const float2 q = *(const float2*)(uv2e + (size_t)nid * D + lane * 2);

        // scores: lane holds s = lane and s = lane + 32
        float sc0 = -1e30f, sc1 = -1e30f;
        const int* seq = node_seq + (size_t)row * S;
        for (int s = 0; s < S; ++s) {
            const int idx = seq[s];
            const float2 kv = *(const float2*)(uv2e + (size_t)idx * D + lane * 2);
            float p = q.x * kv.x + q.y * kv.y;
            #pragma unroll
            for (int off = 16; off > 0; off >>= 1)
                p += __shfl_xor(p, off, 32);
            p *= 0.125f;                       // SCALE = 1/sqrt(64)
            if (lane == s)      sc0 = p;
            if (lane + 32 == s) sc1 = p;
        }

        // softmax over the 50 scores (wave-wide reductions)
        float m = fmaxf(sc0, sc1);
        #pragma unroll
        for (int off = 16; off > 0; off >>= 1)
            m = fmaxf(m, __shfl_xor(m, off, 32));
        const float e0 = __expf(sc0 - m);      // invalid slots: exp(-1e30) -> 0
        const float e1 = __expf(sc1 - m);
        float t = e0 + e1;
        #pragma unroll
        for (int off = 16; off > 0; off >>= 1)
            t += __shfl_xor(t, off, 32);
        const float inv = 1.0f / t;
        const float a0 = e0 * inv;
        const float a1 = e1 * inv;

        // weighted sum of V (== K); re-gather hits WGP$/L2 (table is L2-resident)
        float2 fold = make_float2(0.f, 0.f);
        for (int s = 0; s < S; ++s) {
            const float a = (s < 32) ? __shfl(a0, s, 32) : __shfl(a1, s - 32, 32);
            const int idx = seq[s];
            const float2 kv = *(const float2*)(uv2e + (size_t)idx * D + lane * 2);
            fold.x = fmaf(a, kv.x, fold.x);
            fold.y = fmaf(a, kv.y, fold.y);
        }

        // combined = [self_feats(64) | folded(64)] -> LDS row
        const float2 sf = *(const float2*)(feat + (size_t)nid * D + lane * 2);
        float2* crow = (float2*)(comb + wave * TWO_D);
        crow[lane]      = sf;     // cols [0,64)
        crow[32 + lane] = fold;   // cols [64,128)
    }
    __syncthreads();

    // ---------- Phase 2: out[16,64] = relu(comb[16,128] @ W1^T + b1) ----------
    // waves 0..3 each own one 16-wide N tile; 32 chained f32 WMMAs (K step 4).
    if (wave < 4) {
        const int nt    = wave;
        const int lrow  = lane & 15;          // M row (A) / N col (B,C)
        const int khalf = (lane >> 4) * 2;    // lanes 16-31 hold K+2,K+3
        const float* aBase = comb + lrow * TWO_D + khalf;
        const float* bBase = W1 + (nt * 16 + lrow) * TWO_D + khalf; // B[k,n]=W1[n,k]

        v8f c = {};
        #pragma unroll
        for (int k0 = 0; k0 < TWO_D; k0 += 4) {
            v2f a = *(const v2f*)(aBase + k0);   // ds_load_b64
            v2f b = *(const v2f*)(bBase + k0);   // global_load_b64 (L0/L2 hit)
            c = __builtin_amdgcn_wmma_f32_16x16x4_f32(
                    false, a, false, b, (short)0, c, false, false);
        }

        const float bias = b1[nt * 16 + lrow];
        const int m0 = blockIdx.x * 16 + ((lane >> 4) << 3); // +8 for upper half-wave
        #pragma unroll
        for (int r = 0; r < 8; ++r) {
            const int orow = m0 + r;
            if (orow < B) {
                const float v = c[r] + bias;
                out[(size_t)orow * D + nt * 16 + lrow] = fmaxf(v, 0.f);
            }
        }
    }
}

extern "C" void kernel_launch(void* const* d_in, const int* in_sizes, int n_in,
                              void* d_out, int out_size, void* d_ws, size_t ws_size,
                              hipStream_t stream) {
    const int*   nodes    = (const int*)d_in[0];
    const int*   node_seq = (const int*)d_in[1];
    const float* uv2e     = (const float*)d_in[2];
    const float* feat     = (const float*)d_in[3];
    const float* W1       = (const float*)d_in[4];
    const float* b1       = (const float*)d_in[5];
    float* out = (float*)d_out;

    const int B = in_sizes[0];
    const int S = in_sizes[1] / B;           // 50
    const int blocks = (B + 15) / 16;        // 512 blocks of 16 rows

    folded_encoder_fused<<<blocks, 512, 0, stream>>>(
        nodes, node_seq, uv2e, feat, W1, b1, out, B, S);
}